// Full_Cov_Gaussian_Loss_85014582657567
// MI455X (gfx1250) — compile-verified
//
#include <hip/hip_runtime.h>
#include <hip/hip_bf16.h>
#include <cstddef>

typedef __attribute__((ext_vector_type(2))) float v2f;
typedef __attribute__((ext_vector_type(4))) float v4f;
typedef __attribute__((ext_vector_type(8))) float v8f;

#define BATCH 16
#define NANN  500     // N annotations (background row N..N is weight 0 -> skipped)
#define HW    16384   // 128*128
#define KIDX  512
#define W_COEF 0.01f

// ---------------------------------------------------------------------------
// K1: mw[b,h] = sum_n target[b,n] * m[b,n,h]   (background row has weight 0)
// Bandwidth phase: 525 MB of m streamed once. float4 per lane (b128 loads),
// non-temporal (m is touch-once, > L2), prefetch 8 rows (512KB) ahead.
// Grid: (HW/1024, BATCH), 256 threads; each thread owns 4 consecutive h.
// ---------------------------------------------------------------------------
__global__ void __launch_bounds__(256)
fcg_mw_kernel(const float* __restrict__ m, const float* __restrict__ target,
              float* __restrict__ mw) {
  const int b   = blockIdx.y;
  const int tid = threadIdx.x;
  __shared__ float ann[NANN];
  for (int i = tid; i < NANN; i += 256) ann[i] = target[b * NANN + i];
  __syncthreads();

  const int h = blockIdx.x * 1024 + tid * 4;
  const v4f* mp = reinterpret_cast<const v4f*>(m + (size_t)b * (NANN + 1) * HW + h);
  const int rowStride = HW / 4;  // v4f units between consecutive n rows

  v4f acc = {0.f, 0.f, 0.f, 0.f};
  for (int n = 0; n < NANN; ++n) {
    if (n + 8 < NANN) __builtin_prefetch(mp + 8 * rowStride, 0, 0);
    v4f mv = __builtin_nontemporal_load(mp);
    mp += rowStride;
    const float s = ann[n];
    acc.x += s * mv.x; acc.y += s * mv.y; acc.z += s * mv.z; acc.w += s * mv.w;
  }
  *reinterpret_cast<v4f*>(mw + (size_t)b * HW + h) = acc;
}

// ---------------------------------------------------------------------------
// K2: factor[b] = N / (sum_h mw[b,h] + 1e-12); also f2=factor^2 and zero q[b]
// One block per batch (mw is L2-resident after K1).
// ---------------------------------------------------------------------------
__global__ void __launch_bounds__(256)
fcg_factor_kernel(const float* __restrict__ mw, float* __restrict__ factor,
                  float* __restrict__ f2, float* __restrict__ q) {
  const int b = blockIdx.x, tid = threadIdx.x;
  __shared__ float red[256];
  float s = 0.f;
  for (int h = tid; h < HW; h += 256) s += mw[(size_t)b * HW + h];
  red[tid] = s; __syncthreads();
  for (int st = 128; st > 0; st >>= 1) {
    if (tid < st) red[tid] += red[tid + st];
    __syncthreads();
  }
  if (tid == 0) {
    const float f = (float)NANN / (red[0] + 1e-12f);
    factor[b] = f;
    f2[b] = f * f;
    q[b] = 0.f;   // accumulator for WMMA phase (re-zeroed every call)
  }
}

// ---------------------------------------------------------------------------
// K3: lg1[b] = (sum_h (x-mw*f)^2 / v) / f^2 ; also gather tM[b,k] = tmp[Minds]
// ---------------------------------------------------------------------------
__global__ void __launch_bounds__(256)
fcg_lg1_gather_kernel(const float* __restrict__ mw, const float* __restrict__ x,
                      const float* __restrict__ v, const int* __restrict__ Minds,
                      const float* __restrict__ factor, const float* __restrict__ f2,
                      float* __restrict__ lg1, float* __restrict__ tM) {
  const int b = blockIdx.x, tid = threadIdx.x;
  const float f = factor[b];
  __shared__ float red[256];
  float s = 0.f;
  for (int h = tid; h < HW; h += 256) {
    const float t = x[(size_t)b * HW + h] - mw[(size_t)b * HW + h] * f;
    s += t * t / v[(size_t)b * HW + h];
  }
  red[tid] = s; __syncthreads();
  for (int st = 128; st > 0; st >>= 1) {
    if (tid < st) red[tid] += red[tid + st];
    __syncthreads();
  }
  if (tid == 0) lg1[b] = red[0] / f2[b];

  for (int k = tid; k < KIDX; k += 256) {
    const int i = Minds[b * KIDX + k];
    tM[b * KIDX + k] = x[(size_t)b * HW + i] - mw[(size_t)b * HW + i] * f;
  }
}

// ---------------------------------------------------------------------------
// K5: q[b] += t^T * Bmat[b] * t   via V_WMMA_F32_16X16X4_F32 (wave32).
// One wave per (batch, 16-row tile of Bmat). Per the CDNA5 16x4 f32 A layout:
//   lanes 0-15  hold A[M=lane][K=0..1] in VGPR0..1
//   lanes 16-31 hold A[M=lane-16][K=2..3]
// B (4x16) is t[j0..j0+3] replicated across all 16 columns:
//   lanes 0-15: rows K=0,1 -> {t[j0], t[j0+1]}; lanes 16-31: rows K=2,3.
// Every D column then equals the matvec chunk y = Bmat_tile * t_chunk;
// dot with t, reduce all 32 lanes, divide by 16 (16 identical columns).
// EXEC is all-ones throughout the WMMA loop (blockDim=32, uniform control).
// ---------------------------------------------------------------------------
__global__ void __launch_bounds__(32)
fcg_lg2_wmma_kernel(const float* __restrict__ Bmat, const float* __restrict__ tM,
                    float* __restrict__ q) {
  const int b     = blockIdx.y;
  const int ktile = blockIdx.x;        // 0..31
  const int lane  = threadIdx.x;       // 0..31
  __shared__ float t[KIDX];
  for (int i = lane; i < KIDX; i += 32) t[i] = tM[b * KIDX + i];
  __syncthreads();

  const int k0   = ktile * 16;
  const int row  = k0 + (lane & 15);
  const int jsel = lane >> 4;          // 0: K=0,1 half; 1: K=2,3 half
  const float* Brow = Bmat + ((size_t)b * KIDX + row) * KIDX;

  v8f acc = {0.f, 0.f, 0.f, 0.f, 0.f, 0.f, 0.f, 0.f};
  for (int j0 = 0; j0 < KIDX; j0 += 4) {
    const float2 av = *reinterpret_cast<const float2*>(Brow + j0 + 2 * jsel);
    v2f A; A.x = av.x; A.y = av.y;
    v2f Bv; Bv.x = t[j0 + 2 * jsel]; Bv.y = t[j0 + 2 * jsel + 1];
    acc = __builtin_amdgcn_wmma_f32_16x16x4_f32(
        /*neg_a=*/false, A, /*neg_b=*/false, Bv,
        /*c_mod=*/(short)0, acc, /*reuse_a=*/false, /*reuse_b=*/false);
  }

  // C/D layout: VGPR r holds D[M=r][N=lane] (lanes<16) / D[M=8+r][N=lane-16].
  const int off = 8 * jsel;
  float partial = 0.f;
#pragma unroll
  for (int r = 0; r < 8; ++r) partial += acc[r] * t[k0 + off + r];
  // all 16 columns identical -> sum over 32 lanes counts the tile 16 times
  for (int mlane = 16; mlane >= 1; mlane >>= 1)
    partial += __shfl_xor(partial, mlane, 32);
  if (lane == 0) atomicAdd(&q[b], partial * (1.0f / 16.0f));
}

// ---------------------------------------------------------------------------
// K6: out = mean_b 0.5 * (lg1[b] - q[b]/f2[b]) * W_COEF
// ---------------------------------------------------------------------------
__global__ void __launch_bounds__(32)
fcg_finalize_kernel(const float* __restrict__ lg1, const float* __restrict__ q,
                    const float* __restrict__ f2, float* __restrict__ out) {
  const int lane = threadIdx.x;
  float val = 0.f;
  if (lane < BATCH) val = 0.5f * (lg1[lane] - q[lane] / f2[lane]) * W_COEF;
  for (int mlane = 16; mlane >= 1; mlane >>= 1)
    val += __shfl_xor(val, mlane, 32);
  if (lane == 0) out[0] = val / (float)BATCH;
}

// ---------------------------------------------------------------------------
extern "C" void kernel_launch(void* const* d_in, const int* in_sizes, int n_in,
                              void* d_out, int out_size, void* d_ws, size_t ws_size,
                              hipStream_t stream) {
  (void)in_sizes; (void)n_in; (void)out_size; (void)ws_size;
  const float* m      = (const float*)d_in[0];  // (16, 501, 16384)
  const float* v      = (const float*)d_in[1];  // (16, 1, 16384)
  const float* Bmat   = (const float*)d_in[2];  // (16, 512, 512)
  const float* target = (const float*)d_in[3];  // (16, 500)
  const float* x      = (const float*)d_in[4];  // (16, 1, 128, 128)
  const int*   Minds  = (const int*)d_in[5];    // (16, 512)
  float* out = (float*)d_out;

  // workspace layout (floats): mw | factor | f2 | lg1 | q | tM  (~1.06 MB)
  float* ws     = (float*)d_ws;
  float* mw     = ws;                       // 16*16384
  float* factor = mw + (size_t)BATCH * HW;  // 16
  float* f2     = factor + BATCH;           // 16
  float* lg1    = f2 + BATCH;               // 16
  float* q      = lg1 + BATCH;              // 16
  float* tM     = q + BATCH;                // 16*512

  fcg_mw_kernel<<<dim3(HW / 1024, BATCH), 256, 0, stream>>>(m, target, mw);
  fcg_factor_kernel<<<BATCH, 256, 0, stream>>>(mw, factor, f2, q);
  fcg_lg1_gather_kernel<<<BATCH, 256, 0, stream>>>(mw, x, v, Minds, factor, f2, lg1, tM);
  fcg_lg2_wmma_kernel<<<dim3(KIDX / 16, BATCH), 32, 0, stream>>>(Bmat, tM, q);
  fcg_finalize_kernel<<<1, 32, 0, stream>>>(lg1, q, f2, out);
}